// AvULoss_26542897889844
// MI455X (gfx1250) — compile-verified
//
#include <hip/hip_runtime.h>
#include <math.h>

#ifndef __has_builtin
#define __has_builtin(x) 0
#endif

#define GLOBAL_AS __attribute__((address_space(1)))
#define LDS_AS    __attribute__((address_space(3)))

typedef int v4i __attribute__((vector_size(16)));

namespace {
constexpr int   kC            = 32;                       // classes per row
constexpr int   kThreads      = 128;                      // 4 waves (wave32)
constexpr int   kRowsPerTile  = 128;                      // one row per thread per tile
constexpr int   kRowStrideF   = 36;                       // 144B padded row in LDS (16B aligned)
constexpr int   kTileFloats   = kRowsPerTile * kRowStrideF;   // 4608 floats = 18 KB
constexpr int   kChunksPerTile= kRowsPerTile * (kC * 4 / 16); // 1024 x 16B chunks
constexpr int   kBlocks       = 4096;
constexpr float kEps          = 1e-10f;
}

// ---- CDNA5 async global->LDS copy (ASYNCcnt-tracked), with asm fallback ----
__device__ __forceinline__ void async_g2l_b128(const float* g, float* l) {
#if __has_builtin(__builtin_amdgcn_global_load_async_to_lds_b128)
  __builtin_amdgcn_global_load_async_to_lds_b128(
      (GLOBAL_AS v4i*)g, (LDS_AS v4i*)l, 0, 0);
#else
  unsigned lo = (unsigned)(unsigned long long)(LDS_AS float*)l;
  asm volatile("global_load_async_to_lds_b128 %0, %1, off"
               :: "v"(lo), "v"(g) : "memory");
#endif
}

__device__ __forceinline__ void wait_async0() {
#if __has_builtin(__builtin_amdgcn_s_wait_asynccnt)
  __builtin_amdgcn_s_wait_asynccnt(0);
#else
  asm volatile("s_wait_asynccnt 0" ::: "memory");
#endif
}

__device__ __forceinline__ float wave_sum(float v) {
#pragma unroll
  for (int off = 16; off > 0; off >>= 1) v += __shfl_xor(v, off, 32);
  return v;
}

// Stage 1: per-block partial sums of {n_ac, n_au, n_ic, n_iu}
__global__ __launch_bounds__(kThreads) void avu_partials(
    const float* __restrict__ logits, const int* __restrict__ labels,
    const float* __restrict__ unc_th, float* __restrict__ partials, int nrows) {
  __shared__ __align__(16) float tilebuf[2][kTileFloats];   // ~36 KB double buffer
  __shared__ float wred[kThreads / 32][4];

  const int tid    = threadIdx.x;
  const int ntiles = (nrows + kRowsPerTile - 1) / kRowsPerTile;
  const float uth  = unc_th[0];

  float a0 = 0.f, a1 = 0.f, a2 = 0.f, a3 = 0.f;

  int tile = blockIdx.x;
  int buf  = 0;

  // Prefetch first tile (coalesced: lane L loads the L-th consecutive 16B chunk)
  if (tile < ntiles) {
    const int r0 = tile * kRowsPerTile;
#pragma unroll
    for (int c2 = 0; c2 < kChunksPerTile / kThreads; ++c2) {
      int k = c2 * kThreads + tid;
      int row = k >> 3, j = k & 7;
      int grow = r0 + row;
      if (grow < nrows)
        async_g2l_b128(logits + (size_t)grow * kC + j * 4,
                       &tilebuf[0][row * kRowStrideF + j * 4]);
    }
  }

  for (; tile < ntiles; tile += gridDim.x) {
    wait_async0();          // our wave's async loads for `tile` have landed
    __syncthreads();        // everyone's loads landed + prev compute done

    // Kick off next tile into the other buffer; overlaps with compute below
    const int nexttile = tile + gridDim.x;
    if (nexttile < ntiles) {
      const int r0 = nexttile * kRowsPerTile;
#pragma unroll
      for (int c2 = 0; c2 < kChunksPerTile / kThreads; ++c2) {
        int k = c2 * kThreads + tid;
        int row = k >> 3, j = k & 7;
        int grow = r0 + row;
        if (grow < nrows)
          async_g2l_b128(logits + (size_t)grow * kC + j * 4,
                         &tilebuf[buf ^ 1][row * kRowStrideF + j * 4]);
      }
    }

    const int grow = tile * kRowsPerTile + tid;   // one row per thread
    if (grow < nrows) {
      float x[kC];
      const float4* rp = (const float4*)&tilebuf[buf][tid * kRowStrideF];
#pragma unroll
      for (int j = 0; j < 8; ++j) {
        float4 v = rp[j];
        x[4 * j + 0] = v.x; x[4 * j + 1] = v.y;
        x[4 * j + 2] = v.z; x[4 * j + 3] = v.w;
      }
      // max + first-occurrence argmax
      float m = x[0]; int idx = 0;
#pragma unroll
      for (int j = 1; j < kC; ++j) if (x[j] > m) { m = x[j]; idx = j; }
      // S = sum e^(x-m), T = sum (x-m) e^(x-m)
      float S = 0.f, T = 0.f;
#pragma unroll
      for (int j = 0; j < kC; ++j) {
        float d = x[j] - m;
        float e = __expf(d);
        S += e; T += d * e;
      }
      float conf = 1.0f / S;                 // max softmax prob = e^0 / S
      float unc  = __logf(S) - T * conf;     // predictive entropy
      float t    = tanhf(unc);
      int   lab  = labels[grow];
      bool okc = (lab == idx);
      bool cer = (unc <= uth);
      float base = okc ? conf : (1.0f - conf);
      float w    = cer ? (1.0f - t) : t;
      float contrib = base * w;
      a0 += (okc && cer)   ? contrib : 0.f;  // n_ac
      a1 += (okc && !cer)  ? contrib : 0.f;  // n_au
      a2 += (!okc && cer)  ? contrib : 0.f;  // n_ic
      a3 += (!okc && !cer) ? contrib : 0.f;  // n_iu
    }
    buf ^= 1;
  }

  // wave32 reduce -> LDS -> per-block partial
  a0 = wave_sum(a0); a1 = wave_sum(a1); a2 = wave_sum(a2); a3 = wave_sum(a3);
  const int lane = tid & 31, wid = tid >> 5;
  if (lane == 0) { wred[wid][0] = a0; wred[wid][1] = a1;
                   wred[wid][2] = a2; wred[wid][3] = a3; }
  __syncthreads();
  if (tid == 0) {
    float s0 = 0, s1 = 0, s2 = 0, s3 = 0;
#pragma unroll
    for (int w = 0; w < kThreads / 32; ++w) {
      s0 += wred[w][0]; s1 += wred[w][1]; s2 += wred[w][2]; s3 += wred[w][3];
    }
    ((float4*)partials)[blockIdx.x] = make_float4(s0, s1, s2, s3);
  }
}

// Stage 2: deterministic reduction of partials + final scalar loss
__global__ __launch_bounds__(256) void avu_finalize(
    const float* __restrict__ partials, float* __restrict__ out, int nb) {
  __shared__ float wred[8][4];
  float s0 = 0, s1 = 0, s2 = 0, s3 = 0;
  const float4* p4 = (const float4*)partials;
  for (int i = threadIdx.x; i < nb; i += blockDim.x) {
    float4 v = p4[i];
    s0 += v.x; s1 += v.y; s2 += v.z; s3 += v.w;
  }
  s0 = wave_sum(s0); s1 = wave_sum(s1); s2 = wave_sum(s2); s3 = wave_sum(s3);
  const int lane = threadIdx.x & 31, wid = threadIdx.x >> 5;
  if (lane == 0) { wred[wid][0] = s0; wred[wid][1] = s1;
                   wred[wid][2] = s2; wred[wid][3] = s3; }
  __syncthreads();
  if (threadIdx.x == 0) {
    float ac = 0, au = 0, ic = 0, iu = 0;
#pragma unroll
    for (int w = 0; w < 8; ++w) {
      ac += wred[w][0]; au += wred[w][1]; ic += wred[w][2]; iu += wred[w][3];
    }
    float avu = (ac + iu) / (ac + au + ic + iu + kEps);
    out[0] = -logf(avu + kEps);   // BETA = 1.0
  }
}

extern "C" void kernel_launch(void* const* d_in, const int* in_sizes, int n_in,
                              void* d_out, int out_size, void* d_ws, size_t ws_size,
                              hipStream_t stream) {
  const float* logits = (const float*)d_in[0];
  const int*   labels = (const int*)d_in[1];
  const float* unc_th = (const float*)d_in[2];
  float*       partials = (float*)d_ws;

  const int nrows = in_sizes[0] / kC;

  int nb = kBlocks;
  if (ws_size < (size_t)nb * 4 * sizeof(float)) {
    nb = (int)(ws_size / (4 * sizeof(float)));
    if (nb < 1) nb = 1;
  }

  avu_partials<<<nb, kThreads, 0, stream>>>(logits, labels, unc_th, partials, nrows);
  avu_finalize<<<1, 256, 0, stream>>>(partials, (float*)d_out, nb);
}